// ModelNew_5531917877457
// MI455X (gfx1250) — compile-verified
//
#include <hip/hip_runtime.h>

typedef __attribute__((ext_vector_type(2))) float v2f;
typedef __attribute__((ext_vector_type(8))) float v8f;

#define NMAT 4096
#define BM 256
#define BN 128
#define BK 16

__global__ __launch_bounds__(256) void triu_gemm_f32_wmma(const float* __restrict__ A,
                                                          const float* __restrict__ B,
                                                          float* __restrict__ C) {
    const int bj = blockIdx.x;  // N tile index (128 wide)
    const int bi = blockIdx.y;  // M tile index (256 tall)
    const int t  = threadIdx.x;

    // Tile entirely below the diagonal -> exactly zero, just fill.
    if (bi * BM > bj * BN + (BN - 1)) {
        const float4 z = make_float4(0.f, 0.f, 0.f, 0.f);
        float* base = C + (size_t)bi * BM * NMAT + (size_t)bj * BN;
        for (int s = t; s < (BM * BN / 4); s += 256) {
            int r = s >> 5;            // 32 float4 per 128-wide row
            int c = (s & 31) << 2;
            *(float4*)(base + (size_t)r * NMAT + c) = z;
        }
        return;
    }

    __shared__ float As[BM][BK + 4];   // row stride 20 floats (80B, 16B-aligned for b128 stores)
    __shared__ float Bs[BN][BK + 2];   // transposed [n][k]; stride 18 floats (even -> 8B-aligned b64)

    const int wave  = t >> 5;
    const int lane  = t & 31;
    const int waveM = wave & 3;        // 4 waves along M: 64 rows each
    const int waveN = wave >> 2;       // 2 waves along N: 64 cols each
    const int m16   = lane & 15;
    const int khalf = (lane >> 4) << 1;  // lanes 0-15 -> K+0/1, lanes 16-31 -> K+2/3

    const int row0 = bi * BM;
    const int col0 = bj * BN;
    const int kBeg = row0;             // A triu: cols < row are zero
    const int kEnd = (bj + 1) * BN;    // B triu: rows > col are zero

    v8f acc[4][4];
#pragma unroll
    for (int mt = 0; mt < 4; ++mt)
#pragma unroll
        for (int nt = 0; nt < 4; ++nt)
            acc[mt][nt] = (v8f){0.f, 0.f, 0.f, 0.f, 0.f, 0.f, 0.f, 0.f};

    float4 pa[4];  // A slab prefetch: 256x16 = 1024 float4, 4 per thread
    float4 pb[2];  // B slab prefetch: 16x128 = 512 float4, 2 per thread

    auto stage_load = [&](int kb) {
#pragma unroll
        for (int i = 0; i < 4; ++i) {
            int s = t + 256 * i;
            int ar = s >> 2, ac = (s & 3) << 2;      // 4 float4 per A row
            pa[i] = *(const float4*)(A + (size_t)(row0 + ar) * NMAT + kb + ac);
        }
#pragma unroll
        for (int i = 0; i < 2; ++i) {
            int s = t + 256 * i;
            int br = s >> 5, bc = (s & 31) << 2;     // 32 float4 per B row (row = k)
            pb[i] = *(const float4*)(B + (size_t)(kb + br) * NMAT + col0 + bc);
        }
    };

    stage_load(kBeg);

    for (int kbase = kBeg; kbase < kEnd; kbase += BK) {
        // Commit prefetched slab to LDS (B transposed: [n][k]).
#pragma unroll
        for (int i = 0; i < 4; ++i) {
            int s = t + 256 * i;
            int ar = s >> 2, ac = (s & 3) << 2;
            *(float4*)&As[ar][ac] = pa[i];
        }
#pragma unroll
        for (int i = 0; i < 2; ++i) {
            int s = t + 256 * i;
            int br = s >> 5, bc = (s & 31) << 2;
            Bs[bc + 0][br] = pb[i].x;
            Bs[bc + 1][br] = pb[i].y;
            Bs[bc + 2][br] = pb[i].z;
            Bs[bc + 3][br] = pb[i].w;
        }
        __syncthreads();

        // Kick off next slab's global loads; they overlap the 64 WMMAs below.
        if (kbase + BK < kEnd) stage_load(kbase + BK);

#pragma unroll
        for (int kk = 0; kk < BK; kk += 4) {
            v2f af[4], bf[4];
#pragma unroll
            for (int mt = 0; mt < 4; ++mt)
                af[mt] = *(const v2f*)&As[waveM * 64 + mt * 16 + m16][kk + khalf];
#pragma unroll
            for (int nt = 0; nt < 4; ++nt)
                bf[nt] = *(const v2f*)&Bs[waveN * 64 + nt * 16 + m16][kk + khalf];
#pragma unroll
            for (int mt = 0; mt < 4; ++mt)
#pragma unroll
                for (int nt = 0; nt < 4; ++nt)
                    acc[mt][nt] = __builtin_amdgcn_wmma_f32_16x16x4_f32(
                        false, af[mt], false, bf[nt], (short)0, acc[mt][nt], false, false);
        }
        __syncthreads();
    }

    // Store per the 16x16 f32 C/D layout: VGPR r, lanes 0-15 -> M=r, lanes 16-31 -> M=r+8.
#pragma unroll
    for (int mt = 0; mt < 4; ++mt) {
#pragma unroll
        for (int nt = 0; nt < 4; ++nt) {
            int r0 = row0 + waveM * 64 + mt * 16 + ((lane >> 4) << 3);
            int c  = col0 + waveN * 64 + nt * 16 + m16;
#pragma unroll
            for (int r = 0; r < 8; ++r)
                C[(size_t)(r0 + r) * NMAT + c] = acc[mt][nt][r];
        }
    }
}

extern "C" void kernel_launch(void* const* d_in, const int* in_sizes, int n_in,
                              void* d_out, int out_size, void* d_ws, size_t ws_size,
                              hipStream_t stream) {
    const float* A = (const float*)d_in[0];
    const float* B = (const float*)d_in[1];
    float* C = (float*)d_out;
    dim3 grid(NMAT / BN, NMAT / BM);  // x = N tiles (32), y = M tiles (16)
    triu_gemm_f32_wmma<<<grid, dim3(256), 0, stream>>>(A, B, C);
}